// CRFLayer_17179869849
// MI455X (gfx1250) — compile-verified
//
#include <hip/hip_runtime.h>

// ---------------------------------------------------------------------------
// CRF layer, MI455X (gfx1250, wave32).
// Kernel 1: emissions = x @ W^T + b via native fp32 WMMA (v_wmma_f32_16x16x4_f32)
//           - exact fp32, zero conversion VALU; GEMM is HBM-bound
//             (128MB x + 8MB out @ 23.3 TB/s ~ 6us), instruction issue ~5us.
// Kernel 2: per-batch CRF forward scan (sequential over S) + numerator,
//           transitions staged in LDS, 256 threads = 64 columns x 4 slices.
// Kernel 3: deterministic fixed-order reduction of per-batch partials.
// ---------------------------------------------------------------------------

#define BDIM 64
#define SDIM 512
#define NDIM 1024
#define TDIM 64
#define MROWS (BDIM * SDIM)   // 32768 rows of the emissions GEMM

typedef __attribute__((ext_vector_type(2))) float v2f;
typedef __attribute__((ext_vector_type(8))) float v8f;

static __device__ inline v8f wmma_f32(v2f a, v2f b, v8f c) {
    // 8 args: (neg_a, A, neg_b, B, c_mod, C, reuse_a, reuse_b)
    return __builtin_amdgcn_wmma_f32_16x16x4_f32(
        false, a, false, b, (short)0, c, false, false);
}

// ---------------------------------------------------------------------------
// Kernel 1: emissions[M=32768][T=64] = x @ W^T + b
// One wave handles a 16-row tile x all 64 output columns (4 C tiles).
// fp32 A 16x4 layout: lane<16 -> row=lane, K=0,1; lane>=16 -> row=lane-16, K=2,3.
// fp32 B 4x16 layout mirrors it with col=lane&15; B[k][t] = W[t][k], so both
// fragments are contiguous float2 loads from row-major x / W.
// 4 waves / block, 512 blocks = 2048 row tiles.
// ---------------------------------------------------------------------------
__global__ __launch_bounds__(128) void emis_gemm_kernel(
    const float* __restrict__ x, const float* __restrict__ W,
    const float* __restrict__ bias, float* __restrict__ em) {
    const int lane    = threadIdx.x & 31;
    const int wave    = threadIdx.x >> 5;
    const int gtile   = blockIdx.x * 4 + wave;     // 0 .. 2047
    const int rowBase = gtile * 16;
    const int rsel    = lane & 15;                 // row (A) / col (B) in tile
    const int kb      = (lane >> 4) * 2;           // K sub-offset for this lane

    const float* xrow = x + (size_t)(rowBase + rsel) * NDIM + kb;
    const float* w0   = W + (size_t)( 0 + rsel) * NDIM + kb;
    const float* w1   = W + (size_t)(16 + rsel) * NDIM + kb;
    const float* w2   = W + (size_t)(32 + rsel) * NDIM + kb;
    const float* w3   = W + (size_t)(48 + rsel) * NDIM + kb;

    v8f acc0 = {}, acc1 = {}, acc2 = {}, acc3 = {};

#pragma unroll 8
    for (int k = 0; k < NDIM; k += 4) {
        const v2f a  = *reinterpret_cast<const v2f*>(xrow + k);
        const v2f b0 = *reinterpret_cast<const v2f*>(w0 + k);
        const v2f b1 = *reinterpret_cast<const v2f*>(w1 + k);
        const v2f b2 = *reinterpret_cast<const v2f*>(w2 + k);
        const v2f b3 = *reinterpret_cast<const v2f*>(w3 + k);
        acc0 = wmma_f32(a, b0, acc0);
        acc1 = wmma_f32(a, b1, acc1);
        acc2 = wmma_f32(a, b2, acc2);
        acc3 = wmma_f32(a, b3, acc3);
    }

    // C layout: VGPR r -> M = r (lanes 0-15) / r+8 (lanes 16-31); N = lane&15
    const float bi0 = bias[ 0 + rsel];
    const float bi1 = bias[16 + rsel];
    const float bi2 = bias[32 + rsel];
    const float bi3 = bias[48 + rsel];
#pragma unroll
    for (int r = 0; r < 8; ++r) {
        const int M = rowBase + r + (lane >> 4) * 8;
        float* dst = em + (size_t)M * TDIM;
        dst[ 0 + rsel] = acc0[r] + bi0;
        dst[16 + rsel] = acc1[r] + bi1;
        dst[32 + rsel] = acc2[r] + bi2;
        dst[48 + rsel] = acc3[r] + bi3;
    }
}

// ---------------------------------------------------------------------------
// Kernel 2: per-batch CRF forward scan + numerator.
// grid = 64 batches, block = 256 threads: thread = (q = tid>>6) * 64 + t'.
// Each (q, t') covers 16 source tags; 4 partials merged via LDS.
// ---------------------------------------------------------------------------
__global__ __launch_bounds__(256) void crf_scan_kernel(
    const float* __restrict__ em, const int* __restrict__ y,
    const float* __restrict__ trans, float* __restrict__ partials) {
    __shared__ float sTrans[TDIM * TDIM];   // 16 KB, [t][t'] row-major
    __shared__ float sAlpha[TDIM];
    __shared__ float sM[256];
    __shared__ float sS[256];
    __shared__ float sRed[256];

    const int tid = threadIdx.x;
    const int b   = blockIdx.x;
    const int q   = tid >> 6;       // 0..3  (slice of source tags)
    const int tp  = tid & 63;       // destination tag t'

    const float* erow = em + (size_t)b * SDIM * TDIM;
    const int*   yb   = y + b * SDIM;

    for (int i = tid; i < TDIM * TDIM; i += 256) sTrans[i] = trans[i];
    if (tid < TDIM) sAlpha[tid] = erow[tid];   // alpha = emissions[b, 0, :]

    // numerator partial: sum_s em[s, y[s]] + sum_{s<511} trans[y[s], y[s+1]]
    float np = 0.f;
    for (int s = tid; s < SDIM; s += 256) {
        np += erow[(size_t)s * TDIM + yb[s]];
        if (s < SDIM - 1) np += trans[yb[s] * TDIM + yb[s + 1]];
    }
    sRed[tid] = np;
    __syncthreads();

    float numer = 0.f;
    if (tid == 0) {
        for (int i = 0; i < 256; ++i) numer += sRed[i];  // fixed order
    }

    // sequential forward scan over s = 1..511
    for (int s = 1; s < SDIM; ++s) {
        float m = -1e30f;
#pragma unroll
        for (int i = 0; i < 16; ++i) {
            const int t = q * 16 + i;
            const float v = sAlpha[t] + sTrans[t * TDIM + tp];
            m = fmaxf(m, v);
        }
        float sum = 0.f;
#pragma unroll
        for (int i = 0; i < 16; ++i) {
            const int t = q * 16 + i;
            const float v = sAlpha[t] + sTrans[t * TDIM + tp];
            sum += __expf(v - m);
        }
        sM[q * 64 + tp] = m;
        sS[q * 64 + tp] = sum;
        __syncthreads();
        float na = 0.f;
        if (tid < TDIM) {
            const float m0 = sM[tp], m1 = sM[64 + tp], m2 = sM[128 + tp], m3 = sM[192 + tp];
            const float mm = fmaxf(fmaxf(m0, m1), fmaxf(m2, m3));
            const float ss = sS[tp]       * __expf(m0 - mm)
                           + sS[64 + tp]  * __expf(m1 - mm)
                           + sS[128 + tp] * __expf(m2 - mm)
                           + sS[192 + tp] * __expf(m3 - mm);
            na = mm + __logf(ss) + erow[(size_t)s * TDIM + tp];
        }
        __syncthreads();
        if (tid < TDIM) sAlpha[tp] = na;
        __syncthreads();
    }

    if (tid == 0) {
        float mm = -1e30f;
        for (int t = 0; t < TDIM; ++t) mm = fmaxf(mm, sAlpha[t]);
        float ss = 0.f;
        for (int t = 0; t < TDIM; ++t) ss += __expf(sAlpha[t] - mm);
        const float den = mm + __logf(ss);
        partials[b] = den - numer;
    }
}

// ---------------------------------------------------------------------------
// Kernel 3: deterministic fixed-order sum of per-batch partials.
// ---------------------------------------------------------------------------
__global__ void finalize_kernel(const float* __restrict__ partials,
                                float* __restrict__ out) {
    if (threadIdx.x == 0 && blockIdx.x == 0) {
        float s = 0.f;
        for (int i = 0; i < BDIM; ++i) s += partials[i];
        out[0] = s;
    }
}

extern "C" void kernel_launch(void* const* d_in, const int* in_sizes, int n_in,
                              void* d_out, int out_size, void* d_ws, size_t ws_size,
                              hipStream_t stream) {
    const float* x     = (const float*)d_in[0];   // (64, 512, 1024) f32
    const int*   y     = (const int*)  d_in[1];   // (64, 512) i32
    const float* W     = (const float*)d_in[2];   // (64, 1024) f32
    const float* bias  = (const float*)d_in[3];   // (64,) f32
    const float* trans = (const float*)d_in[4];   // (64, 64) f32

    float* em       = (float*)d_ws;                                   // 8 MB
    float* partials = (float*)((char*)d_ws + (size_t)MROWS * TDIM * sizeof(float));

    emis_gemm_kernel<<<MROWS / 16 / 4, 128, 0, stream>>>(x, W, bias, em);
    crf_scan_kernel<<<BDIM, 256, 0, stream>>>(em, y, trans, partials);
    finalize_kernel<<<1, 32, 0, stream>>>(partials, (float*)d_out);
}